// GeoLayer_2388001817296
// MI455X (gfx1250) — compile-verified
//
#include <hip/hip_runtime.h>

#define NN 50000
#define NE 800000
#define ET (NE + NN)      // edges + self loops
#define HEADS 4
#define OC 32
#define C128 128          // IN_CH == HEADS*OUT_CH == 128

typedef __attribute__((ext_vector_type(2))) float v2f;
typedef __attribute__((ext_vector_type(8))) float v8f;

// ---------------- init: out = bias (broadcast), amax = -inf, esum = 0 ----------
__global__ void k_init(float4* __restrict__ out4, const float4* __restrict__ bias4,
                       float* __restrict__ amax, float* __restrict__ esum) {
    long i = (long)blockIdx.x * blockDim.x + threadIdx.x;
    if (i < (long)NN * (C128 / 4)) out4[i] = bias4[i & 31];
    if (i < (long)NN * HEADS) { amax[i] = -3.0e38f; esum[i] = 0.0f; }
}

// ---------------- x = feat @ weight via V_WMMA_F32_16X16X4_F32 ------------------
// block = 256 threads = 8 waves; block -> 16-row strip, wave w -> cols [16w,16w+16)
__global__ void k_gemm(const float* __restrict__ feat, const float* __restrict__ w,
                       float* __restrict__ x) {
    const int wave  = threadIdx.x >> 5;
    const int lane  = threadIdx.x & 31;
    const int lm    = lane & 15;          // row (A) / col (B,D) within tile
    const int khalf = (lane >> 4) * 2;    // lanes 0-15 hold K{0,1}, 16-31 hold K{2,3}
    const long mbase = (long)blockIdx.x * 16;
    const int  nbase = wave * 16;

    const float* arow = feat + (mbase + lm) * C128;
    v8f acc = {};
#pragma unroll
    for (int k = 0; k < C128; k += 4) {
        v2f a, b;
        a.x = arow[k + khalf];
        a.y = arow[k + khalf + 1];
        b.x = w[(k + khalf)     * C128 + nbase + lm];
        b.y = w[(k + khalf + 1) * C128 + nbase + lm];
        acc = __builtin_amdgcn_wmma_f32_16x16x4_f32(false, a, false, b,
                                                    (short)0, acc, false, false);
    }
#pragma unroll
    for (int r = 0; r < 8; ++r) {
        long m = mbase + r + 8 * (lane >> 4);   // D: VGPR r -> row r (+8 for hi lanes)
        x[m * C128 + nbase + lm] = acc[r];
    }
}

// ---------------- per-node attention dots: ai/aj[n,h] ---------------------------
// one 128-thread block per node; each wave32 == one head (wave32 on gfx1250)
__global__ void k_nodeatt(const float* __restrict__ x, const float* __restrict__ att,
                          float* __restrict__ ai, float* __restrict__ aj) {
    int n = blockIdx.x;
    int t = threadIdx.x;        // 0..127
    int h = t >> 5;
    int c = t & 31;
    float v  = x[(long)n * C128 + t];
    float r1 = v * att[h * (2 * OC) + c];        // att_i (dst half)
    float r2 = v * att[h * (2 * OC) + OC + c];   // att_j (src half)
#pragma unroll
    for (int off = 16; off; off >>= 1) {
        r1 += __shfl_down(r1, off, 32);
        r2 += __shfl_down(r2, off, 32);
    }
    if (c == 0) { ai[n * HEADS + h] = r1; aj[n * HEADS + h] = r2; }
}

__device__ __forceinline__ void atomic_max_f32(float* addr, float val) {
    unsigned* ua  = (unsigned*)addr;
    unsigned  cur = __float_as_uint(*addr);
    while (__uint_as_float(cur) < val) {
        unsigned prev = atomicCAS(ua, cur, __float_as_uint(val));
        if (prev == cur) break;
        cur = prev;
    }
}

__device__ __forceinline__ void edge_sd(long e, const int* __restrict__ ei,
                                        int& s, int& d) {
    if (e < NE) { s = ei[e]; d = ei[NE + e]; }
    else        { s = d = (int)(e - NE); }      // appended self-loops
}

// ---------------- pass A: alpha = leakyrelu(ai[dst]+aj[src]); segment max -------
__global__ void k_alpha(const int* __restrict__ ei, const float* __restrict__ ai,
                        const float* __restrict__ aj, float* __restrict__ alpha,
                        float* __restrict__ amax) {
    long i = (long)blockIdx.x * blockDim.x + threadIdx.x;
    if (i >= (long)ET * HEADS) return;
    long e = i >> 2; int h = (int)(i & 3);
    int s, d; edge_sd(e, ei, s, d);
    float a = ai[d * HEADS + h] + aj[s * HEADS + h];
    a = (a > 0.0f) ? a : 0.2f * a;              // leaky_relu, slope 0.2
    alpha[i] = a;
    atomic_max_f32(&amax[s * HEADS + h], a);
}

// ---------------- pass B: ex = exp(alpha - amax[src]); segment sum --------------
__global__ void k_expsum(const int* __restrict__ ei, const float* __restrict__ amax,
                         float* __restrict__ alpha, float* __restrict__ esum) {
    long i = (long)blockIdx.x * blockDim.x + threadIdx.x;
    if (i >= (long)ET * HEADS) return;
    long e = i >> 2; int h = (int)(i & 3);
    int s, d; edge_sd(e, ei, s, d);
    float ex = __expf(alpha[i] - amax[s * HEADS + h]);
    alpha[i] = ex;
    atomicAdd(&esum[s * HEADS + h], ex);
}

// ---------------- pass C: out[dst] += x[src] * ex/(esum[src]+eps) ---------------
// 32 threads per edge, each owns one float4 -> global_load_b128 for x[src];
// x (25.6MB) and out (25.6MB) are L2-resident (192MB L2) so atomics stay on-chip
__global__ void k_aggregate(const int* __restrict__ ei, const float* __restrict__ x,
                            const float* __restrict__ alpha, const float* __restrict__ esum,
                            float* __restrict__ out) {
    long i = (long)blockIdx.x * blockDim.x + threadIdx.x;
    if (i >= (long)ET * 32) return;
    long e = i >> 5;
    int  q = (int)(i & 31);     // which float4 (8 per head)
    int  h = q >> 3;
    int s, d; edge_sd(e, ei, s, d);
    float w = alpha[e * HEADS + h] / (esum[s * HEADS + h] + 1e-16f);
    float4 xv = ((const float4*)(x + (long)s * C128))[q];
    float* o = out + (long)d * C128 + q * 4;
    atomicAdd(o + 0, xv.x * w);
    atomicAdd(o + 1, xv.y * w);
    atomicAdd(o + 2, xv.z * w);
    atomicAdd(o + 3, xv.w * w);
}

extern "C" void kernel_launch(void* const* d_in, const int* in_sizes, int n_in,
                              void* d_out, int out_size, void* d_ws, size_t ws_size,
                              hipStream_t stream) {
    const float* feat   = (const float*)d_in[0];
    const int*   ei     = (const int*)  d_in[1];   // [2, NE] flat: src then dst
    const float* weight = (const float*)d_in[2];   // [128, 128]
    const float* att    = (const float*)d_in[3];   // [1, 4, 64]
    const float* bias   = (const float*)d_in[4];   // [128]
    float* out = (float*)d_out;

    float* ws    = (float*)d_ws;
    float* x     = ws;                              // NN*128
    float* ai    = x    + (long)NN * C128;          // NN*4
    float* aj    = ai   + (long)NN * HEADS;         // NN*4
    float* amax  = aj   + (long)NN * HEADS;         // NN*4
    float* esum  = amax + (long)NN * HEADS;         // NN*4
    float* alpha = esum + (long)NN * HEADS;         // ET*4

    long n_init = (long)NN * (C128 / 4);
    k_init<<<(int)((n_init + 255) / 256), 256, 0, stream>>>(
        (float4*)out, (const float4*)bias, amax, esum);

    k_gemm<<<NN / 16, 256, 0, stream>>>(feat, weight, x);

    k_nodeatt<<<NN, 128, 0, stream>>>(x, att, ai, aj);

    long n_eh = (long)ET * HEADS;
    k_alpha <<<(int)((n_eh + 255) / 256), 256, 0, stream>>>(ei, ai, aj, alpha, amax);
    k_expsum<<<(int)((n_eh + 255) / 256), 256, 0, stream>>>(ei, amax, alpha, esum);

    long n_agg = (long)ET * 32;
    k_aggregate<<<(int)((n_agg + 255) / 256), 256, 0, stream>>>(ei, x, alpha, esum, out);
}